// GCNGraphClassifier_33457795236077
// MI455X (gfx1250) — compile-verified
//
#include <hip/hip_runtime.h>

#define NN 150000
#define NE 2400000
#define NG 1024
#define FD 32
#define EPSV 1e-5f

typedef __attribute__((ext_vector_type(2))) float v2f;
typedef __attribute__((ext_vector_type(8))) float v8f;

__global__ void k_zero(float* __restrict__ p, int n) {
  int i = blockIdx.x * blockDim.x + threadIdx.x;
  if (i < n) p[i] = 0.0f;
}

__global__ void k_deg(const int* __restrict__ dst, float* __restrict__ deg, int e_cnt) {
  int e = blockIdx.x * blockDim.x + threadIdx.x;
  if (e < e_cnt) atomicAdd(&deg[dst[e]], 1.0f);
}

__global__ void k_counts(const int* __restrict__ batch, float* __restrict__ counts, int n) {
  int i = blockIdx.x * blockDim.x + threadIdx.x;
  if (i < n) atomicAdd(&counts[batch[i]], 1.0f);
}

__global__ void k_dinv(float* __restrict__ deg_io, int n) {
  int i = blockIdx.x * blockDim.x + threadIdx.x;
  if (i < n) deg_io[i] = rsqrtf(deg_io[i] + 1.0f);
}

// H[16-row tile, 32 cols] = X[16 x K] @ W[K x 32] via V_WMMA_F32_16X16X4_F32.
// One wave per 16-row tile; K stepped by 4 (K=3 zero-padded). EXEC is all-ones
// for every participating wave (wave-uniform guard, blockDim % 32 == 0).
template <int K>
__global__ void k_transform_wmma(const float* __restrict__ X, const float* __restrict__ W,
                                 float* __restrict__ H, int nwaves) {
  int gtid = blockIdx.x * blockDim.x + threadIdx.x;
  int wave = gtid >> 5;
  if (wave >= nwaves) return;
  int lane = threadIdx.x & 31;
  int m = lane & 15;   // row within tile (A) / col within tile (B, C/D)
  int hi = lane >> 4;  // K sub-pair select for A/B; row-half select for C/D
  int row = wave * 16 + m;

  v8f acc0 = {};
  v8f acc1 = {};
#pragma unroll
  for (int k = 0; k < K; k += 4) {
    int ka = k + 2 * hi;
    // A 16x4 fragment: lanes 0-15 hold K=k,k+1; lanes 16-31 hold K=k+2,k+3
    v2f a;
    a.x = (ka + 0 < K) ? X[row * K + ka + 0] : 0.0f;
    a.y = (ka + 1 < K) ? X[row * K + ka + 1] : 0.0f;
    // B 4x16 fragments for column tiles [0..15] and [16..31]
    v2f b0, b1;
    b0.x = (ka + 0 < K) ? W[(ka + 0) * FD + m] : 0.0f;
    b0.y = (ka + 1 < K) ? W[(ka + 1) * FD + m] : 0.0f;
    b1.x = (ka + 0 < K) ? W[(ka + 0) * FD + 16 + m] : 0.0f;
    b1.y = (ka + 1 < K) ? W[(ka + 1) * FD + 16 + m] : 0.0f;
    acc0 = __builtin_amdgcn_wmma_f32_16x16x4_f32(false, a, false, b0, (short)0, acc0, false, false);
    acc1 = __builtin_amdgcn_wmma_f32_16x16x4_f32(false, a, false, b1, (short)0, acc1, false, false);
  }
  // C/D layout: VGPR v -> rows v (lanes 0-15) and v+8 (lanes 16-31), col = lane%16
  int r0 = wave * 16 + 8 * hi;
#pragma unroll
  for (int v = 0; v < 8; ++v) {
    H[(r0 + v) * FD + m] = acc0[v];
    H[(r0 + v) * FD + 16 + m] = acc1[v];
  }
}

// 8 threads per edge, 4 feats each: agg[dst] += coef * h[src]
__global__ void k_scatter(const int* __restrict__ src, const int* __restrict__ dst,
                          const float* __restrict__ dinv, const float* __restrict__ H,
                          float* __restrict__ AGG, int e_cnt) {
  int t = blockIdx.x * blockDim.x + threadIdx.x;
  int e = t >> 3;
  int sub = t & 7;
  if (e >= e_cnt) return;
  int s = src[e];
  int d = dst[e];
  int e2 = e + 512;
  if (e2 < e_cnt) {
    // warm L2/WGP$ for a future source row (global_prefetch_b8)
    __builtin_prefetch(&H[(long)src[e2] * FD + sub * 4], 0, 1);
  }
  float coef = dinv[s] * dinv[d];
  float4 hv = *reinterpret_cast<const float4*>(H + (long)s * FD + sub * 4);
  float* out = AGG + (long)d * FD + sub * 4;
  atomicAdd(out + 0, hv.x * coef);
  atomicAdd(out + 1, hv.y * coef);
  atomicAdd(out + 2, hv.z * coef);
  atomicAdd(out + 3, hv.w * coef);
}

// agg += dinv^2 * h + bias ; accumulate per-graph feature sums
__global__ void k_combine(const float* __restrict__ H, const float* __restrict__ dinv,
                          const float* __restrict__ bias, const int* __restrict__ batch,
                          float* __restrict__ AGG, float* __restrict__ gsum, int n) {
  int i = blockIdx.x * blockDim.x + threadIdx.x;
  if (i >= n) return;
  int node = i >> 5;
  int f = i & 31;
  float di = dinv[node];
  float val = AGG[i] + di * di * H[i] + bias[f];
  AGG[i] = val;
  atomicAdd(&gsum[batch[node] * FD + f], val);
}

// out = val - mean_scale * mean ; accumulate per-graph variance sums
__global__ void k_center(const float* __restrict__ gsum, const float* __restrict__ counts,
                         const float* __restrict__ ms, const int* __restrict__ batch,
                         float* __restrict__ AGG, float* __restrict__ gvar, int n) {
  int i = blockIdx.x * blockDim.x + threadIdx.x;
  if (i >= n) return;
  int node = i >> 5;
  int f = i & 31;
  int g = batch[node];
  float mean = gsum[g * FD + f] / counts[g];
  float out = AGG[i] - ms[f] * mean;
  AGG[i] = out;
  atomicAdd(&gvar[g * FD + f], out * out);
}

// y = relu(w*centered/std + b [+ residual]) ; optional mean-pool accumulate
__global__ void k_finalize(const float* __restrict__ gvar, const float* __restrict__ counts,
                           const float* __restrict__ w, const float* __restrict__ b,
                           const int* __restrict__ batch, const float* __restrict__ AGG,
                           const float* __restrict__ Xres, float* __restrict__ Xout,
                           float* __restrict__ pool, int n) {
  int i = blockIdx.x * blockDim.x + threadIdx.x;
  if (i >= n) return;
  int node = i >> 5;
  int f = i & 31;
  int g = batch[node];
  float var = gvar[g * FD + f] / counts[g];
  float stdv = sqrtf(var + EPSV);
  float y = w[f] * AGG[i] / stdv + b[f];
  if (Xres) y += Xres[i];
  y = fmaxf(y, 0.0f);
  Xout[i] = y;
  if (pool) atomicAdd(&pool[g * FD + f], y);
}

__global__ void k_linear(const float* __restrict__ pool, const float* __restrict__ counts,
                         const float* __restrict__ lw, const float* __restrict__ lb,
                         float* __restrict__ out) {
  int i = blockIdx.x * blockDim.x + threadIdx.x;
  if (i >= NG * 3) return;
  int g = i / 3;
  int j = i % 3;
  float inv = 1.0f / counts[g];
  float acc = lb[j];
#pragma unroll
  for (int f = 0; f < FD; ++f) acc += pool[g * FD + f] * inv * lw[f * 3 + j];
  out[i] = acc;
}

extern "C" void kernel_launch(void* const* d_in, const int* in_sizes, int n_in,
                              void* d_out, int out_size, void* d_ws, size_t ws_size,
                              hipStream_t stream) {
  const float* x     = (const float*)d_in[0];
  const int*   edge  = (const int*)d_in[1];
  const int*   batch = (const int*)d_in[2];
  const float* W1 = (const float*)d_in[3],  *b1 = (const float*)d_in[4];
  const float* g1w = (const float*)d_in[5], *g1b = (const float*)d_in[6], *g1m = (const float*)d_in[7];
  const float* W2 = (const float*)d_in[8],  *b2 = (const float*)d_in[9];
  const float* g2w = (const float*)d_in[10], *g2b = (const float*)d_in[11], *g2m = (const float*)d_in[12];
  const float* W3 = (const float*)d_in[13], *b3 = (const float*)d_in[14];
  const float* g3w = (const float*)d_in[15], *g3b = (const float*)d_in[16], *g3m = (const float*)d_in[17];
  const float* lw = (const float*)d_in[18], *lb = (const float*)d_in[19];
  const int* src = edge;
  const int* dst = edge + NE;

  char* ws = (char*)d_ws;
  size_t off = 0;
  auto alloc = [&](size_t elems) {
    float* p = (float*)(ws + off);
    off += (elems * sizeof(float) + 255) & ~(size_t)255;
    return p;
  };
  float* dinv   = alloc(NN);
  float* counts = alloc(NG);
  float* H      = alloc((size_t)NN * FD);
  float* AGG    = alloc((size_t)NN * FD);
  float* XA     = alloc((size_t)NN * FD);
  float* XB     = alloc((size_t)NN * FD);
  float* gsum   = alloc((size_t)NG * FD);
  float* gvar   = alloc((size_t)NG * FD);
  float* pool   = alloc((size_t)NG * FD);
  (void)ws_size; (void)n_in; (void)in_sizes; (void)out_size;

  const int B = 256;
  const int NF = NN * FD;
  const int GF = NG * FD;
  const int NWAVES = NN / 16;               // 9375, exact
  const int TR_THREADS = NWAVES * 32;

  // degree, counts, dinv_sqrt
  k_zero<<<(NN + B - 1) / B, B, 0, stream>>>(dinv, NN);
  k_zero<<<(NG + B - 1) / B, B, 0, stream>>>(counts, NG);
  k_zero<<<(GF + B - 1) / B, B, 0, stream>>>(pool, GF);
  k_deg<<<(NE + B - 1) / B, B, 0, stream>>>(dst, dinv, NE);
  k_counts<<<(NN + B - 1) / B, B, 0, stream>>>(batch, counts, NN);
  k_dinv<<<(NN + B - 1) / B, B, 0, stream>>>(dinv, NN);

  auto run_layer = [&](const float* Xin, int K, const float* W, const float* bias,
                       const float* gw, const float* gb, const float* gms,
                       const float* Xres, float* Xout, float* poolp) {
    if (K == 3)
      k_transform_wmma<3><<<(TR_THREADS + B - 1) / B, B, 0, stream>>>(Xin, W, H, NWAVES);
    else
      k_transform_wmma<32><<<(TR_THREADS + B - 1) / B, B, 0, stream>>>(Xin, W, H, NWAVES);
    k_zero<<<(NF + B - 1) / B, B, 0, stream>>>(AGG, NF);
    k_zero<<<(GF + B - 1) / B, B, 0, stream>>>(gsum, GF);
    k_zero<<<(GF + B - 1) / B, B, 0, stream>>>(gvar, GF);
    k_scatter<<<((NE * 8) + B - 1) / B, B, 0, stream>>>(src, dst, dinv, H, AGG, NE);
    k_combine<<<(NF + B - 1) / B, B, 0, stream>>>(H, dinv, bias, batch, AGG, gsum, NF);
    k_center<<<(NF + B - 1) / B, B, 0, stream>>>(gsum, counts, gms, batch, AGG, gvar, NF);
    k_finalize<<<(NF + B - 1) / B, B, 0, stream>>>(gvar, counts, gw, gb, batch, AGG, Xres, Xout, poolp, NF);
  };

  run_layer(x,  3,  W1, b1, g1w, g1b, g1m, nullptr, XA, nullptr);
  run_layer(XA, 32, W2, b2, g2w, g2b, g2m, XA,      XB, nullptr);
  run_layer(XB, 32, W3, b3, g3w, g3b, g3m, XB,      XA, pool);

  k_linear<<<(NG * 3 + B - 1) / B, B, 0, stream>>>(pool, counts, lw, lb, (float*)d_out);
}